// PatchSlicedTransport_34926674051676
// MI455X (gfx1250) — compile-verified
//
#include <hip/hip_runtime.h>
#include <math.h>

// Problem constants (from reference)
#define NH_    8
#define NWID   8
#define NKER   64          // patch blocks
#define NSUB   48          // 4*4*3 patch dims == K columns per block
#define NDIM   3072        // 32*32*3
#define NTR    3072        // NKER * NSUB
#define MKNOT  200
#define NBATCH 8192
#define QSTR   49          // padded LDS stride for Q (48x48 block)
#define WSTR   49          // padded LDS stride for per-wave transpose buffer

typedef __attribute__((ext_vector_type(2))) float v2f;
typedef __attribute__((ext_vector_type(8))) float v8f;

// D = A(16x4,f32) * B(4x16,f32) + C(16x16,f32), CDNA5 WMMA.
// A frag:  lane l, vgpr v -> A[l%16][2*(l/16)+v]
// B frag:  lane l, vgpr v -> B[2*(l/16)+v][l%16]
// C/D:     lane l, vgpr v -> D[v+8*(l/16)][l%16]
__device__ __forceinline__ v8f wmma_f32_16x16x4(v2f a, v2f b, v8f c) {
#if __has_builtin(__builtin_amdgcn_wmma_f32_16x16x4_f32)
  return __builtin_amdgcn_wmma_f32_16x16x4_f32(false, a, false, b, (short)0, c,
                                               false, false);
#else
  // cross-lane emulation fallback (keeps semantics if builtin is absent)
  int lane = (int)(threadIdx.x & 31u);
  int l16 = lane & 15;
#pragma unroll
  for (int v = 0; v < 8; ++v) {
    int m = v + 8 * (lane >> 4);
    float sum = 0.f;
#pragma unroll
    for (int k = 0; k < 4; ++k) {
      int srcA = m + 16 * (k >> 1);
      int srcB = l16 + 16 * (k >> 1);
      float av = __shfl((k & 1) ? a.y : a.x, srcA, 32);
      float bv = __shfl((k & 1) ? b.y : b.x, srcB, 32);
      sum += av * bv;
    }
    c[v] += sum;
  }
  return c;
#endif
}

// Flattened pixel index of element i of patch-block kk, including the (2,2) roll.
__device__ __forceinline__ int pix_index(int kk, int i) {
  int ph = kk >> 3, pw = kk & 7;
  int t = i / 3;
  int c = i - 3 * t;
  int kw = t & 3, kh = t >> 2;
  int h = (ph * 4 + kh + 2) & 31;
  int w = (pw * 4 + kw + 2) & 31;
  return (h * 32 + w) * 3 + c;
}

// ---------------------------------------------------------------------------
// Kernel 0: per-block QR via modified Gram-Schmidt. 64 blocks of 48x48.
// Positive norms reproduce the reference's sign(diag(R)) fix.
// ---------------------------------------------------------------------------
__global__ __launch_bounds__(64) void qr_kernel(const float* __restrict__ Araw,
                                                float* __restrict__ Q) {
  __shared__ float qj[NSUB];
  const int kk = blockIdx.x;
  const int t = threadIdx.x;  // thread t owns column t (t < 48)
  float col[NSUB];
  if (t < NSUB) {
#pragma unroll
    for (int i = 0; i < NSUB; ++i) col[i] = Araw[kk * (NSUB * NSUB) + i * NSUB + t];
  }
  for (int j = 0; j < NSUB; ++j) {
    if (t == j) {
      float s = 0.f;
#pragma unroll
      for (int i = 0; i < NSUB; ++i) s += col[i] * col[i];
      float inv = 1.0f / sqrtf(s);
#pragma unroll
      for (int i = 0; i < NSUB; ++i) { col[i] *= inv; qj[i] = col[i]; }
    }
    __syncthreads();
    if (t < NSUB && t > j) {
      float d = 0.f;
#pragma unroll
      for (int i = 0; i < NSUB; ++i) d += qj[i] * col[i];
#pragma unroll
      for (int i = 0; i < NSUB; ++i) col[i] -= d * qj[i];
    }
    __syncthreads();
  }
  if (t < NSUB) {
#pragma unroll
    for (int i = 0; i < NSUB; ++i) Q[kk * (NSUB * NSUB) + i * NSUB + t] = col[i];
  }
}

// ---------------------------------------------------------------------------
// Kernel 1: build spline knot tables xx, yy (cumsum of exp) and delta (exp).
// One thread per transform row. 3072 rows x 200 knots.
// ---------------------------------------------------------------------------
__global__ __launch_bounds__(256) void knots_kernel(
    const float* __restrict__ x0, const float* __restrict__ logdx,
    const float* __restrict__ y0, const float* __restrict__ logdy,
    const float* __restrict__ logderiv, float* __restrict__ xx,
    float* __restrict__ yy, float* __restrict__ dd) {
  int tr = blockIdx.x * blockDim.x + threadIdx.x;
  if (tr >= NTR) return;
  float cx = x0[tr];
  xx[(size_t)tr * MKNOT] = cx;
  for (int m = 1; m < MKNOT; ++m) {
    cx += __expf(logdx[(size_t)tr * (MKNOT - 1) + m - 1]);
    xx[(size_t)tr * MKNOT + m] = cx;
  }
  float cy = y0[tr];
  yy[(size_t)tr * MKNOT] = cy;
  for (int m = 1; m < MKNOT; ++m) {
    cy += __expf(logdy[(size_t)tr * (MKNOT - 1) + m - 1]);
    yy[(size_t)tr * MKNOT + m] = cy;
  }
  for (int m = 0; m < MKNOT; ++m)
    dd[(size_t)tr * MKNOT + m] = __expf(logderiv[(size_t)tr * MKNOT + m]);
}

// ---------------------------------------------------------------------------
// Kernel 2 (fused): per (block kk, 1024-sample chunk):
//   data0 = gather(data) @ Q            (WMMA f32 16x16x4)
//   y, logd = RQ-spline(data0)          (register D-layout, LDS knot search)
//   out    = data + (y - data0) @ Qt    (WMMA f32 16x16x4, scatter to patch)
//   partial[kk][n] = sum_j logd         (deterministic, no atomics)
// ---------------------------------------------------------------------------
__global__ __launch_bounds__(128) void fused_kernel(
    const float* __restrict__ data, const float* __restrict__ Q,
    const float* __restrict__ xx, const float* __restrict__ yy,
    const float* __restrict__ dd, float* __restrict__ out,
    float* __restrict__ partial) {
  __shared__ float Qs[NSUB * QSTR];      // 48x48 Q block, padded
  __shared__ float Xs[NSUB * MKNOT];     // xx knots for this block's 48 rows
  __shared__ float Wb[4 * 16 * WSTR];    // per-wave (y-data0) transpose buffers

  const int kk = blockIdx.x & 63;
  const int chunk = blockIdx.x >> 6;     // 0..7, each = 1024 samples
  const int tid = threadIdx.x;
  const int lane = tid & 31;
  const int wave = tid >> 5;
  const int l16 = lane & 15;
  const int lh = lane >> 4;

  // Stage Q (row-major i x j) and xx rows kk*48..kk*48+47.
  for (int p = tid; p < NSUB * NSUB; p += 128) {
    int i = p / NSUB, j = p - i * NSUB;
    Qs[i * QSTR + j] = Q[kk * (NSUB * NSUB) + p];
  }
  for (int p = tid; p < NSUB * MKNOT; p += 128)
    Xs[p] = xx[(size_t)kk * NSUB * MKNOT + p];
  __syncthreads();

  // Per-lane gather indices: GEMM1 A-frag pixels and epilogue pixels.
  int pixA[24];
#pragma unroll
  for (int s = 0; s < 12; ++s) {
    pixA[2 * s]     = pix_index(kk, 4 * s + 2 * lh);
    pixA[2 * s + 1] = pix_index(kk, 4 * s + 2 * lh + 1);
  }
  int pixE[3];
#pragma unroll
  for (int it = 0; it < 3; ++it) pixE[it] = pix_index(kk, it * 16 + l16);

  float* mywb = &Wb[wave * 16 * WSTR];

  for (int tt = 0; tt < 16; ++tt) {
    const int tile = chunk * 64 + wave * 16 + tt;  // global 16-sample tile
    const int nbase = tile * 16;
    const float* arow = data + (size_t)(nbase + l16) * NDIM;

    if (tt + 1 < 16)  // prefetch next tile's gather row (global_prefetch_b8)
      __builtin_prefetch(data + (size_t)(nbase + 16 + l16) * NDIM + pixA[0], 0, 3);

    // ---- GEMM1: data0 tile = X(16x48) @ Q(48x48) ----
    v8f acc[3] = {v8f{}, v8f{}, v8f{}};
#pragma unroll
    for (int s = 0; s < 12; ++s) {
      const int ka = 4 * s + 2 * lh;
      v2f a;
      a.x = arow[pixA[2 * s]];
      a.y = arow[pixA[2 * s + 1]];
#pragma unroll
      for (int jt = 0; jt < 3; ++jt) {
        v2f b;
        b.x = Qs[ka * QSTR + jt * 16 + l16];
        b.y = Qs[(ka + 1) * QSTR + jt * 16 + l16];
        acc[jt] = wmma_f32_16x16x4(a, b, acc[jt]);
      }
    }

    // ---- RQ spline on the accumulators (D layout), y-data0 in place ----
    float lacc[8];
#pragma unroll
    for (int v = 0; v < 8; ++v) lacc[v] = 0.f;
#pragma unroll
    for (int jt = 0; jt < 3; ++jt) {
      const int trl = jt * 16 + l16;                 // local transform row 0..47
      const float* xrow = &Xs[trl * MKNOT];
      const float* yrow = yy + ((size_t)(kk * NSUB + trl)) * MKNOT;
      const float* drow = dd + ((size_t)(kk * NSUB + trl)) * MKNOT;
#pragma unroll
      for (int v = 0; v < 8; ++v) {
        const float xv = acc[jt][v];
        // searchsorted (side='left'): first m with xrow[m] >= xv
        int lo = 0, hi = MKNOT;
        while (lo < hi) {
          int mid = (lo + hi) >> 1;
          if (xrow[mid] < xv) lo = mid + 1; else hi = mid;
        }
        float yv, dv;
        if (lo == 0) {
          float d0 = drow[0];
          yv = yrow[0] + d0 * (xv - xrow[0]);
          dv = d0;
        } else if (lo == MKNOT) {
          float dl = drow[MKNOT - 1];
          yv = yrow[MKNOT - 1] + dl * (xv - xrow[MKNOT - 1]);
          dv = dl;
        } else {
          int k = lo - 1;
          float xK = xrow[k], xK1 = xrow[k + 1];
          float yK = yrow[k], yK1 = yrow[k + 1];
          float dK = drow[k], dK1 = drow[k + 1];
          float invw = 1.0f / (xK1 - xK);
          float xi = (xv - xK) * invw;
          xi = fminf(fmaxf(xi, 0.f), 1.f);
          float s = (yK1 - yK) * invw;
          float xi1 = xi * (1.f - xi);
          float den = s + (dK1 + dK - 2.f * s) * xi1;
          float invden = 1.0f / den;
          yv = yK + (yK1 - yK) * (s * xi * xi + dK * xi1) * invden;
          float omxi = 1.f - xi;
          dv = s * s * (dK1 * xi * xi + 2.f * s * xi1 + dK * omxi * omxi) *
               invden * invden;
        }
        acc[jt][v] = yv - xv;     // (y - data0)
        lacc[v] += logf(dv);
      }
    }

    // log-jacobian partials: reduce over the 16 lanes of each half-wave.
#pragma unroll
    for (int v = 0; v < 8; ++v) {
#pragma unroll
      for (int off = 1; off < 16; off <<= 1)
        lacc[v] += __shfl_xor(lacc[v], off, 32);
    }
    if (l16 == 0) {
#pragma unroll
      for (int v = 0; v < 8; ++v)
        partial[(size_t)kk * NBATCH + nbase + v + 8 * lh] = lacc[v];
    }

    // ---- transpose (y - data0) tile into A-fragment layout via LDS ----
#pragma unroll
    for (int jt = 0; jt < 3; ++jt)
#pragma unroll
      for (int v = 0; v < 8; ++v)
        mywb[(v + 8 * lh) * WSTR + jt * 16 + l16] = acc[jt][v];
    __syncthreads();

    // ---- GEMM2: contribution = (y - data0)(16x48) @ Qt(48x48) ----
    v8f o[3] = {v8f{}, v8f{}, v8f{}};
#pragma unroll
    for (int s = 0; s < 12; ++s) {
      const int ka = 4 * s + 2 * lh;
      v2f a;
      a.x = mywb[l16 * WSTR + ka];
      a.y = mywb[l16 * WSTR + ka + 1];
#pragma unroll
      for (int it = 0; it < 3; ++it) {
        v2f b;  // B[k][i] = Q[i][k]
        b.x = Qs[(it * 16 + l16) * QSTR + ka];
        b.y = Qs[(it * 16 + l16) * QSTR + ka + 1];
        o[it] = wmma_f32_16x16x4(a, b, o[it]);
      }
    }

    // ---- epilogue: out[n][pix] = data[n][pix] + contribution ----
#pragma unroll
    for (int it = 0; it < 3; ++it) {
#pragma unroll
      for (int v = 0; v < 8; ++v) {
        const int n = nbase + v + 8 * lh;
        const int r = pixE[it];
        out[(size_t)n * NDIM + r] = data[(size_t)n * NDIM + r] + o[it][v];
      }
    }
    __syncthreads();  // protect per-wave Wb before next tile
  }
}

// ---------------------------------------------------------------------------
// Kernel 3: logj[n] = sum over 64 blocks of partial[kk][n]  (deterministic)
// ---------------------------------------------------------------------------
__global__ __launch_bounds__(256) void logj_kernel(const float* __restrict__ partial,
                                                   float* __restrict__ logj) {
  int n = blockIdx.x * blockDim.x + threadIdx.x;
  if (n >= NBATCH) return;
  float s = 0.f;
#pragma unroll
  for (int kk = 0; kk < NKER; ++kk) s += partial[(size_t)kk * NBATCH + n];
  logj[n] = s;
}

extern "C" void kernel_launch(void* const* d_in, const int* in_sizes, int n_in,
                              void* d_out, int out_size, void* d_ws, size_t ws_size,
                              hipStream_t stream) {
  const float* data     = (const float*)d_in[0];
  const float* A_raw    = (const float*)d_in[1];
  const float* x0       = (const float*)d_in[2];
  const float* logdx    = (const float*)d_in[3];
  const float* y0       = (const float*)d_in[4];
  const float* logdy    = (const float*)d_in[5];
  const float* logderiv = (const float*)d_in[6];

  float* out  = (float*)d_out;                       // (N, NDIM)
  float* logj = out + (size_t)NBATCH * NDIM;         // (N,)

  // Workspace layout (floats): Q | xx | yy | dd | partial   (~10 MB total)
  float* w  = (float*)d_ws;
  float* Qw = w;                                     // 64*48*48
  float* xxw = Qw + NKER * NSUB * NSUB;              // 3072*200
  float* yyw = xxw + (size_t)NTR * MKNOT;
  float* ddw = yyw + (size_t)NTR * MKNOT;
  float* partialw = ddw + (size_t)NTR * MKNOT;       // 64*8192

  qr_kernel<<<dim3(NKER), dim3(64), 0, stream>>>(A_raw, Qw);
  knots_kernel<<<dim3((NTR + 255) / 256), dim3(256), 0, stream>>>(
      x0, logdx, y0, logdy, logderiv, xxw, yyw, ddw);
  fused_kernel<<<dim3(NKER * 8), dim3(128), 0, stream>>>(
      data, Qw, xxw, yyw, ddw, out, partialw);
  logj_kernel<<<dim3((NBATCH + 255) / 256), dim3(256), 0, stream>>>(partialw, logj);
}